// Coarse2FineCharCRFDependency_33904471834771
// MI455X (gfx1250) — compile-verified
//
#include <hip/hip_runtime.h>
#include <hip/hip_bf16.h>
#include <stdint.h>

#define MINV (-1e32f)
#define NTHREADS 256
#define NWAVES 8
#define MAXL 128

// ---------------------------------------------------------------------------
// CDNA5 async global->LDS gather of one f32 per active lane.
// vdst = per-lane LDS byte address, vaddr = per-lane 64-bit global address.
// Tracked by ASYNCcnt; completed via s_wait_asynccnt 0.
// ---------------------------------------------------------------------------
__device__ __forceinline__ void async_gather_f32(const float* gsrc, float* ldst) {
  unsigned lds = (unsigned)(uintptr_t)ldst;  // low 32 bits of flat addr == LDS offset
  unsigned long long ga = (unsigned long long)(uintptr_t)gsrc;
  asm volatile("global_load_async_to_lds_b32 %0, %1, off"
               :: "v"(lds), "v"(ga) : "memory");
}

__device__ __forceinline__ void wait_async_lds() {
  asm volatile("s_wait_asynccnt 0" ::: "memory");
}

// Wave-wide logsumexp over up to 4 terms per lane (128 split points max).
// Invalid slots hold -inf and contribute exp(-inf)=0. Global max is finite
// whenever at least one term exists (terms are >= -2e32).
__device__ __forceinline__ float wave_lse4(float t0, float t1, float t2, float t3) {
  float m = fmaxf(fmaxf(t0, t1), fmaxf(t2, t3));
#pragma unroll
  for (int off = 16; off > 0; off >>= 1)
    m = fmaxf(m, __shfl_xor(m, off, 32));
  float s = __expf(t0 - m) + __expf(t1 - m) + __expf(t2 - m) + __expf(t3 - m);
#pragma unroll
  for (int off = 16; off > 0; off >>= 1)
    s += __shfl_xor(s, off, 32);
  return __logf(s) + m;
}

// ---------------------------------------------------------------------------
// One workgroup per batch element. Both Eisner charts live in LDS, padded to
// stride L+1 so row reads and stride-(L+1) column reads are bank-conflict-free.
// ---------------------------------------------------------------------------
__global__ __launch_bounds__(NTHREADS)
void eisner_inside_kernel(const float* __restrict__ scores,
                          const unsigned char* __restrict__ mask,
                          const int* __restrict__ target,
                          float* __restrict__ part, int L) {
  __shared__ float sc[MAXL * (MAXL + 1)];   // complete-span chart
  __shared__ float si[MAXL * (MAXL + 1)];   // incomplete-span chart
  __shared__ float arcA[2 * MAXL];          // scores[b, i, i+w]  (double-buffered)
  __shared__ float arcB[2 * MAXL];          // scores[b, i+w, i]
  __shared__ float wsum[NWAVES];
  __shared__ int s_len;

  const int tid  = threadIdx.x;
  const int lane = tid & 31;
  const int wv   = tid >> 5;
  const int b    = blockIdx.x;
  const int stride = L + 1;
  const float* S = scores + (size_t)b * L * L;
  const float NEG = -__builtin_inff();

  // init charts to MIN
  for (int x = tid; x < L * stride; x += NTHREADS) { sc[x] = MINV; si[x] = MINV; }
  if (tid == 0) s_len = 0;
  __syncthreads();
  // width-0 complete spans
  for (int d = tid; d < L; d += NTHREADS) sc[d * stride + d] = 0.0f;
  // sentence length (mask popcount) -- integer atomic: deterministic
  {
    int c = 0;
    for (int l = tid; l < L; l += NTHREADS) c += mask[(size_t)b * L + l] ? 1 : 0;
    atomicAdd(&s_len, c);
  }
  // async prefetch of the w=1 score diagonals into buffer 1
  {
    const int w = 1, n = L - w, par = w & 1;
    if (tid < n) {
      async_gather_f32(S + (size_t)tid * L + (tid + w), &arcA[par * MAXL + tid]);
    } else if (tid >= NTHREADS / 2 && (tid - NTHREADS / 2) < n) {
      const int t = tid - NTHREADS / 2;
      async_gather_f32(S + (size_t)(t + w) * L + t, &arcB[par * MAXL + t]);
    }
  }
  wait_async_lds();
  __syncthreads();
  const int len = s_len;

  for (int w = 1; w < L; ++w) {
    const int n = L - w;
    // kick off the DMA for next width's diagonals (overlaps with this width)
    if (w + 1 < L) {
      const int wn = w + 1, nn = L - wn, par = wn & 1;
      if (tid < nn) {
        async_gather_f32(S + (size_t)tid * L + (tid + wn), &arcA[par * MAXL + tid]);
      } else if (tid >= NTHREADS / 2 && (tid - NTHREADS / 2) < nn) {
        const int t = tid - NTHREADS / 2;
        async_gather_f32(S + (size_t)(t + wn) * L + t, &arcB[par * MAXL + t]);
      }
    }
    const float* aA = &arcA[(w & 1) * MAXL];
    const float* aB = &arcB[(w & 1) * MAXL];

    // ---- phase 1: incomplete spans of width w ----
    // il[i] = lse_k( s_c[i, i+k] + s_c[i+w, i+1+k] ), k = 0..w-1
    for (int i = wv; i < n; i += NWAVES) {
      const float* r0 = sc + i * stride + i;             // s_c[i, i+k]
      const float* r1 = sc + (i + w) * stride + (i + 1); // s_c[i+w, i+1+k]
      float t0 = NEG, t1 = NEG, t2 = NEG, t3 = NEG;
      int k = lane;
      if (k < w) t0 = r0[k] + r1[k];
      k += 32; if (k < w) t1 = r0[k] + r1[k];
      k += 32; if (k < w) t2 = r0[k] + r1[k];
      k += 32; if (k < w) t3 = r0[k] + r1[k];
      const float il = wave_lse4(t0, t1, t2, t3);
      if (lane == 0) {
        si[(i + w) * stride + i]  = il + aA[i];   // s_i[i+w, i] + arc(i+w <- i head at i? = scores[b,i,i+w])
        si[i * stride + (i + w)]  = il + aB[i];   // s_i[i, i+w] + scores[b,i+w,i]
      }
    }
    __syncthreads();

    // ---- phase 2: complete spans of width w ----
    // cl[i] = lse_k( s_c[i+k, i]   + s_i[i+w, i+k]   )
    // cr[i] = lse_k( s_i[i, i+1+k] + s_c[i+1+k, i+w] )
    for (int i = wv; i < n; i += NWAVES) {
      const float* colL = sc + i * stride + i;             // s_c[i+k, i]   = colL[k*stride]
      const float* rowI = si + (i + w) * stride + i;       // s_i[i+w, i+k] = rowI[k]
      const float* rowR = si + i * stride + (i + 1);       // s_i[i, i+1+k] = rowR[k]
      const float* colR = sc + (i + 1) * stride + (i + w); // s_c[i+1+k, i+w] = colR[k*stride]
      float c0 = NEG, c1 = NEG, c2 = NEG, c3 = NEG;
      float d0 = NEG, d1 = NEG, d2 = NEG, d3 = NEG;
      int k = lane;
      if (k < w) { c0 = colL[k * stride] + rowI[k]; d0 = rowR[k] + colR[k * stride]; }
      k += 32;
      if (k < w) { c1 = colL[k * stride] + rowI[k]; d1 = rowR[k] + colR[k * stride]; }
      k += 32;
      if (k < w) { c2 = colL[k * stride] + rowI[k]; d2 = rowR[k] + colR[k * stride]; }
      k += 32;
      if (k < w) { c3 = colL[k * stride] + rowI[k]; d3 = rowR[k] + colR[k * stride]; }
      const float clv = wave_lse4(c0, c1, c2, c3);
      float       crv = wave_lse4(d0, d1, d2, d3);
      if (lane == 0) {
        sc[(i + w) * stride + i] = clv;
        if (i == 0 && w != len) crv = MINV;   // single-root constraint
        sc[i * stride + (i + w)] = crv;
      }
    }
    wait_async_lds();   // next width's arc diagonals have landed in LDS
    __syncthreads();    // publish s_c writes + arc buffers
  }

  // gold arc score for this batch element (deterministic tree reduction)
  float g = 0.0f;
  for (int l = tid; l < L; l += NTHREADS) {
    if (mask[(size_t)b * L + l]) {
      const int tg = target[(size_t)b * L + l];
      g += S[(size_t)l * L + tg];
    }
  }
#pragma unroll
  for (int off = 16; off > 0; off >>= 1) g += __shfl_xor(g, off, 32);
  if (lane == 0) wsum[wv] = g;
  __syncthreads();
  if (tid == 0) {
    float gs = 0.0f;
#pragma unroll
    for (int j = 0; j < NWAVES; ++j) gs += wsum[j];
    part[b * 3 + 0] = sc[0 * stride + len];  // logZ_b = s_c[0, len]
    part[b * 3 + 1] = gs;                    // gold score_b
    part[b * 3 + 2] = (float)len;            // token count_b
  }
}

// Single-block deterministic final reduction: loss = (logZ - score) / norm
__global__ __launch_bounds__(NTHREADS)
void finalize_kernel(const float* __restrict__ part, float* __restrict__ out, int B) {
  __shared__ float sz[NWAVES], sg[NWAVES], sn[NWAVES];
  const int tid = threadIdx.x, lane = tid & 31, wv = tid >> 5;
  float z = 0.0f, g = 0.0f, nm = 0.0f;
  for (int i = tid; i < B; i += NTHREADS) {
    z  += part[i * 3 + 0];
    g  += part[i * 3 + 1];
    nm += part[i * 3 + 2];
  }
#pragma unroll
  for (int off = 16; off > 0; off >>= 1) {
    z  += __shfl_xor(z, off, 32);
    g  += __shfl_xor(g, off, 32);
    nm += __shfl_xor(nm, off, 32);
  }
  if (lane == 0) { sz[wv] = z; sg[wv] = g; sn[wv] = nm; }
  __syncthreads();
  if (tid == 0) {
    float Z = 0.0f, G = 0.0f, N = 0.0f;
#pragma unroll
    for (int j = 0; j < NWAVES; ++j) { Z += sz[j]; G += sg[j]; N += sn[j]; }
    out[0] = (Z - G) / N;
  }
}

extern "C" void kernel_launch(void* const* d_in, const int* in_sizes, int n_in,
                              void* d_out, int out_size, void* d_ws, size_t ws_size,
                              hipStream_t stream) {
  (void)n_in; (void)out_size; (void)ws_size;
  const float*         scores = (const float*)d_in[0];
  const unsigned char* mask   = (const unsigned char*)d_in[1];
  const int*           target = (const int*)d_in[2];

  const int BLL = in_sizes[0];      // B*L*L
  const int BL  = in_sizes[1];      // B*L
  const int L   = BLL / BL;         // 128
  const int B   = BL / L;           // 256
  if (L > MAXL) return;

  float* part = (float*)d_ws;       // B * 3 floats of per-block partials

  eisner_inside_kernel<<<B, NTHREADS, 0, stream>>>(scores, mask, target, part, L);
  finalize_kernel<<<1, NTHREADS, 0, stream>>>(part, (float*)d_out, B);
}